// FastViTReparamLKConv_23252952941048
// MI455X (gfx1250) — compile-verified
//
#include <hip/hip_runtime.h>
#include <math.h>

// ---------------------------------------------------------------------------
// FastViT reparam large-kernel depthwise block for MI455X (gfx1250).
// x:(32,384,64,64) f32 -> dwconv7x7 s2 p3 -> BN(batch stats) +
//                         dwconv3x3 s2 p1 -> BN(batch stats) -> exact GELU
// Memory-bound (~6 FLOP/B, ~240MB mandatory traffic @23.3TB/s ~ 10us).
// CDNA5 path: TDM tensor_load_to_lds stages each 64x64 channel plane into a
// zero-padded LDS tile in ONE DMA per block, using the TDM pad feature
// (64 dwords data + 8 dwords skip == our 72-dword row pitch). Stencil uses
// register row-reuse; BN stats need a global reduction, hence 3 passes.
// ---------------------------------------------------------------------------

#define C_CH   384
#define NBATCH 32
#define HIN    64
#define HOUT   32
#define TROWS  70           // lds tile rows: iy in [-3,66] -> row = iy+3
#define TSTRIDE 72          // lds tile row stride (floats); data cols 4..67
#define ROFF   3
#define COFF   4            // col = ix + 4 keeps 16B alignment of data region
#define EPSV   1e-5f

typedef __attribute__((ext_vector_type(4))) unsigned int v4u;
typedef __attribute__((ext_vector_type(8))) int          v8i;
typedef __attribute__((ext_vector_type(4))) int          v4i;

#if defined(__has_builtin)
#if __has_builtin(__builtin_amdgcn_tensor_load_to_lds)
#define USE_TDM 1
#endif
#endif

// Fallback: per-lane CDNA5 async copy (ASYNCcnt-tracked).
__device__ __forceinline__ void async_b128_to_lds(void* lds_dst, const void* gsrc) {
  uint32_t l = (uint32_t)(uintptr_t)lds_dst;
  asm volatile("global_load_async_to_lds_b128 %0, %1, off"
               :: "v"(l), "v"(gsrc) : "memory");
}
__device__ __forceinline__ void wait_async0() {
  asm volatile("s_wait_asynccnt 0" ::: "memory");
}

// Stage the 64x64 f32 plane into tile[] at rows 3..66, cols 4..67.
__device__ __forceinline__ void stage_tile(float* tile, const float* plane, int t) {
#ifdef USE_TDM
  if (t < 32) {  // wave 0 issues the DMA; TDM ignores EXEC, waves 1..7 skip
    uint64_t ga = (uint64_t)(uintptr_t)plane;                       // global byte addr
    uint32_t la = (uint32_t)(uintptr_t)tile + (ROFF * TSTRIDE + COFF) * 4; // 880B, 16B aligned
    // D# group0: count=1 | lds_addr | global_addr[56:0] | type=2
    v4u g0 = { 1u,
               la,
               (unsigned int)(ga & 0xffffffffu),
               (unsigned int)((ga >> 32) & 0x01ffffffu) | (2u << 30) };
    // D# group1: data_size=4B(2)<<16 | pad_enable<<20 | pad_interval=5(64dw)<<22
    //            | pad_amount=7(8dw)<<25 ; tensor_dim0=64 ; tensor_dim1=64 ;
    //            tile_dim0=64 ; tile_dim1=64 ; dim0_stride=64 ; 2-D tile.
    v8i g1 = { (int)((2u << 16) | (1u << 20) | (5u << 22) | (7u << 25)),
               (int)(64u << 16),   // tensor_dim0[15:0] in bits[63:48]
               (int)(64u << 16),   // tensor_dim0[31:16]=0 ; tensor_dim1[15:0]=64
               (int)(64u << 16),   // tensor_dim1[31:16]=0 ; tile_dim0=64
               64,                 // tile_dim1=64 ; tile_dim2=0 (2-D)
               64,                 // tensor_dim0_stride[31:0]=64
               0, 0 };
    v4i gz = { 0, 0, 0, 0 };       // groups 2/3: unused for 2-D tile
#if defined(__clang_major__) && (__clang_major__ >= 23)
    v8i gz8 = { 0, 0, 0, 0, 0, 0, 0, 0 };
    __builtin_amdgcn_tensor_load_to_lds(g0, g1, gz, gz, gz8, 0);
#else
    __builtin_amdgcn_tensor_load_to_lds(g0, g1, gz, gz, 0);
#endif
    __builtin_amdgcn_s_wait_tensorcnt(0);
  }
#else
#pragma unroll
  for (int j = 0; j < 4; ++j) {
    int q = t + j * 256;
    int row = q >> 4, col4 = q & 15;
    async_b128_to_lds(&tile[(row + ROFF) * TSTRIDE + COFF + col4 * 4],
                      plane + row * HIN + col4 * 4);
  }
  wait_async0();
#endif
}

template <bool WRITE_OUT>
__global__ __launch_bounds__(256) void fastvit_dwconv_kernel(
    const float* __restrict__ x,
    const float* __restrict__ w_lk,   // [384,49]
    const float* __restrict__ w_sk,   // [384,9]
    float* __restrict__ stats,        // [384,4]  (pass A accumulate)
    const float* __restrict__ params, // [384,4]  (pass C: a7,b7,a3,b3)
    float* __restrict__ out)
{
  __shared__ float tile[TROWS * TSTRIDE];
  __shared__ float red[8 * 4];

  const int t   = threadIdx.x;
  const int bid = blockIdx.x;           // 0 .. 32*384-1
  const int c   = bid % C_CH;
  const int n   = bid / C_CH;
  const float* plane = x + (size_t)(n * C_CH + c) * (HIN * HIN);

  // ---- zero the halo (TDM pad region is skipped-not-written => disjoint) ----
  for (int i = t; i < TROWS * TSTRIDE; i += 256) {
    int row = i / TSTRIDE;
    int col = i - row * TSTRIDE;
    bool halo = (row < ROFF) | (row >= HIN + ROFF) | (col < COFF) | (col >= HIN + COFF);
    if (halo) tile[i] = 0.0f;
  }

  // ---- DMA the plane into LDS (one tensor op per block) ----
  stage_tile(tile, plane, t);

  const float* __restrict__ w7 = w_lk + c * 49;   // block-uniform -> s_loads
  const float* __restrict__ w3 = w_sk + c * 9;

  __syncthreads();    // TDM done (wave0 waited TENSORcnt) + halo zeros visible

  // ---- stencil: each thread owns 4 vertically stacked outputs ----
  const int ox  = t & 31;
  const int oy0 = (t >> 5) * 4;                   // 0,4,...,28
  const int col7 = 2 * ox + 1;                    // ix = 2ox-3+kx -> col = ix+COFF
  const int col3 = 2 * ox + 3;
  const int rb   = 2 * oy0;

  float a7[4] = {0.f, 0.f, 0.f, 0.f};
  float a3[4] = {0.f, 0.f, 0.f, 0.f};

  // 7x7: lds row = 2*oy + ky = rb + (2j + ky); rel r in 0..13, rows reused across j
#pragma unroll
  for (int r = 0; r < 14; ++r) {
    float v[7];
    const float* src = &tile[(rb + r) * TSTRIDE + col7];
#pragma unroll
    for (int k = 0; k < 7; ++k) v[k] = src[k];
#pragma unroll
    for (int j = 0; j < 4; ++j) {
      int ky = r - 2 * j;
      if (ky >= 0 && ky < 7) {
#pragma unroll
        for (int k = 0; k < 7; ++k) a7[j] = fmaf(v[k], w7[ky * 7 + k], a7[j]);
      }
    }
  }
  // 3x3: lds row = rb + 2 + (2j + ky); rel r in 0..8
#pragma unroll
  for (int r = 0; r < 9; ++r) {
    float v[3];
    const float* src = &tile[(rb + 2 + r) * TSTRIDE + col3];
#pragma unroll
    for (int k = 0; k < 3; ++k) v[k] = src[k];
#pragma unroll
    for (int j = 0; j < 4; ++j) {
      int ky = r - 2 * j;
      if (ky >= 0 && ky < 3) {
#pragma unroll
        for (int k = 0; k < 3; ++k) a3[j] = fmaf(v[k], w3[ky * 3 + k], a3[j]);
      }
    }
  }

  if constexpr (WRITE_OUT) {
    // fused BN affine + sum + exact GELU
    const float s7 = params[c * 4 + 0], b7 = params[c * 4 + 1];
    const float s3 = params[c * 4 + 2], b3 = params[c * 4 + 3];
    float* oplane = out + (size_t)(n * C_CH + c) * (HOUT * HOUT);
#pragma unroll
    for (int j = 0; j < 4; ++j) {
      float y = fmaf(a7[j], s7, b7) + fmaf(a3[j], s3, b3);
      float g = 0.5f * y * (1.0f + erff(y * 0.7071067811865476f));
      oplane[(oy0 + j) * HOUT + ox] = g;
    }
  } else {
    // per-channel sum / sum-of-squares for BN statistics
    float s7 = 0.f, q7 = 0.f, s3 = 0.f, q3 = 0.f;
#pragma unroll
    for (int j = 0; j < 4; ++j) {
      s7 += a7[j]; q7 = fmaf(a7[j], a7[j], q7);
      s3 += a3[j]; q3 = fmaf(a3[j], a3[j], q3);
    }
#pragma unroll
    for (int off = 16; off > 0; off >>= 1) {   // wave32 butterfly
      s7 += __shfl_xor(s7, off, 32);
      q7 += __shfl_xor(q7, off, 32);
      s3 += __shfl_xor(s3, off, 32);
      q3 += __shfl_xor(q3, off, 32);
    }
    const int lane = t & 31, w = t >> 5;
    if (lane == 0) {
      red[w * 4 + 0] = s7; red[w * 4 + 1] = q7;
      red[w * 4 + 2] = s3; red[w * 4 + 3] = q3;
    }
    __syncthreads();
    if (t < 4) {
      float acc = 0.f;
#pragma unroll
      for (int w8 = 0; w8 < 8; ++w8) acc += red[w8 * 4 + t];
      unsafeAtomicAdd(&stats[c * 4 + t], acc);   // global_atomic_add_f32
    }
  }
}

__global__ void fastvit_bn_params_kernel(
    const float* __restrict__ stats,
    const float* __restrict__ gamma7, const float* __restrict__ beta7,
    const float* __restrict__ gamma3, const float* __restrict__ beta3,
    float* __restrict__ params)
{
  int c = blockIdx.x * blockDim.x + threadIdx.x;
  if (c >= C_CH) return;
  const float inv_n = 1.0f / (float)(NBATCH * HOUT * HOUT);
  float m7 = stats[c * 4 + 0] * inv_n;
  float v7 = fmaxf(stats[c * 4 + 1] * inv_n - m7 * m7, 0.0f);
  float m3 = stats[c * 4 + 2] * inv_n;
  float v3 = fmaxf(stats[c * 4 + 3] * inv_n - m3 * m3, 0.0f);
  float sc7 = gamma7[c] * rsqrtf(v7 + EPSV);
  float sc3 = gamma3[c] * rsqrtf(v3 + EPSV);
  params[c * 4 + 0] = sc7;
  params[c * 4 + 1] = beta7[c] - m7 * sc7;
  params[c * 4 + 2] = sc3;
  params[c * 4 + 3] = beta3[c] - m3 * sc3;
}

extern "C" void kernel_launch(void* const* d_in, const int* in_sizes, int n_in,
                              void* d_out, int out_size, void* d_ws, size_t ws_size,
                              hipStream_t stream) {
  (void)in_sizes; (void)n_in; (void)out_size; (void)ws_size;
  const float* x        = (const float*)d_in[0];
  const float* w_lk     = (const float*)d_in[1];
  const float* gamma_lk = (const float*)d_in[2];
  const float* beta_lk  = (const float*)d_in[3];
  const float* w_sk     = (const float*)d_in[4];
  const float* gamma_sk = (const float*)d_in[5];
  const float* beta_sk  = (const float*)d_in[6];
  float* out = (float*)d_out;

  float* stats  = (float*)d_ws;           // [384,4]
  float* params = stats + C_CH * 4;       // [384,4]

  hipMemsetAsync(stats, 0, C_CH * 4 * sizeof(float), stream);  // capture-safe

  dim3 grid(NBATCH * C_CH), block(256);
  // Pass A: conv both kernels, accumulate per-channel sum / sumsq
  fastvit_dwconv_kernel<false><<<grid, block, 0, stream>>>(
      x, w_lk, w_sk, stats, nullptr, nullptr);
  // Pass B: fold stats into per-channel scale/bias
  fastvit_bn_params_kernel<<<1, C_CH, 0, stream>>>(
      stats, gamma_lk, beta_lk, gamma_sk, beta_sk, params);
  // Pass C: recompute conv, fused BN + add + exact GELU
  fastvit_dwconv_kernel<true><<<grid, block, 0, stream>>>(
      x, w_lk, w_sk, nullptr, params, out);
}